// Heirachical_Loss_42030549958856
// MI455X (gfx1250) — compile-verified
//
#include <hip/hip_runtime.h>
#include <hip/hip_bf16.h>

// Hierarchical softmax loss on MI455X (gfx1250).
// B=2048 rows, L=10000 leaves, K=10, D=4. Balanced tree => internal node
// probs are contiguous segment sums of leaf probs. Memory-bound:
// 81.9 MB @ 23.3 TB/s ~= 3.5 us floor. Strategy: one workgroup per row,
// async-DMA the 40KB row into LDS (CDNA5 GLOBAL_LOAD_ASYNC_TO_LDS path,
// ASYNCcnt), then do max / exp / 4-level tree sums entirely from LDS.

#define BB        2048
#define LL        10000
#define KK        10
#define DD        4
#define NTHREADS  256
#define NWAVES    (NTHREADS / 32)
#define L1N       1000
#define L2N       100
#define L3N       10

// Pointer types for the CDNA5 async-to-LDS builtin (param 0 is a v4i pointer
// in AS1 per clang's diagnostic; AS3 for the LDS side).
typedef int v4i_ __attribute__((vector_size(16)));
typedef __attribute__((address_space(1))) v4i_* gbl_v4i_p;
typedef __attribute__((address_space(3))) v4i_* lds_v4i_p;

__device__ __forceinline__ float wave_max32(float v) {
#pragma unroll
  for (int off = 16; off; off >>= 1)
    v = fmaxf(v, __shfl_xor(v, off, 32));
  return v;
}

__device__ __forceinline__ float wave_sum32(float v) {
#pragma unroll
  for (int off = 16; off; off >>= 1)
    v += __shfl_xor(v, off, 32);
  return v;
}

extern "C" __global__ void __launch_bounds__(NTHREADS)
hloss_row_kernel(const float* __restrict__ outputs,
                 const int*   __restrict__ target,
                 float*       __restrict__ out,      // [0]=loss, [1..B]=preds, [B+1]=dist
                 float*       __restrict__ ws_loss,  // [B]
                 float*       __restrict__ ws_dist)  // [B]
{
  __shared__ __align__(16) float s_row[LL];   // 40000 B staged row
  __shared__ float s_l1[L1N];                 // level-1 segment sums (width 10)
  __shared__ float s_l2[L2N];                 // level-2 (width 100)
  __shared__ float s_l3[L3N];                 // level-3 (width 1000)
  __shared__ float s_red[NWAVES];
  __shared__ float s_max;

  const int row = blockIdx.x;
  const int tid = threadIdx.x;
  const float* __restrict__ grow = outputs + (size_t)row * LL;

  // ---- Stage row into LDS via CDNA5 async DMA (ASYNCcnt path) ----
#if __has_builtin(__builtin_amdgcn_global_load_async_to_lds_b128)
  for (int i = tid; i < LL / 4; i += NTHREADS) {
    __builtin_amdgcn_global_load_async_to_lds_b128(
        (gbl_v4i_p)(grow + 4 * i),
        (lds_v4i_p)(s_row + 4 * i),
        0, 0);
  }
#if __has_builtin(__builtin_amdgcn_s_wait_asynccnt)
  __builtin_amdgcn_s_wait_asynccnt(0);
#else
  asm volatile("s_wait_asynccnt 0x0" ::: "memory");
#endif
#else
  // Fallback: vectorized copy through VGPRs.
  for (int i = tid; i < LL / 4; i += NTHREADS) {
    reinterpret_cast<float4*>(s_row)[i] =
        reinterpret_cast<const float4*>(grow)[i];
  }
#endif
  __syncthreads();

  // ---- Pass 1: row max, vectorized b128 LDS reads (LL/4 = 2500 float4s) ----
  float m = -__builtin_inff();
  {
    const float4* r4 = reinterpret_cast<const float4*>(s_row);
    for (int i = tid; i < LL / 4; i += NTHREADS) {
      float4 v = r4[i];
      m = fmaxf(m, fmaxf(fmaxf(v.x, v.y), fmaxf(v.z, v.w)));
    }
  }
  m = wave_max32(m);
  if ((tid & 31) == 0) s_red[tid >> 5] = m;
  __syncthreads();
  if (tid == 0) {
    float mm = s_red[0];
#pragma unroll
    for (int w = 1; w < NWAVES; ++w) mm = fmaxf(mm, s_red[w]);
    s_max = mm;
  }
  __syncthreads();
  const float rmax = s_max;

  // ---- Pass 2: exp + level-1 segment sums (one node per thread => deterministic).
  // Node base = n*40 B, 8-B aligned => 5x b64 LDS reads per node.
  for (int n = tid; n < L1N; n += NTHREADS) {
    const float2* p2 = reinterpret_cast<const float2*>(s_row + n * KK);
    float s = 0.f;
#pragma unroll
    for (int j = 0; j < KK / 2; ++j) {
      float2 v = p2[j];
      s += expf(v.x - rmax) + expf(v.y - rmax);
    }
    s_l1[n] = s;
  }
  __syncthreads();

  if (tid < L2N) {
    float s = 0.f;
#pragma unroll
    for (int j = 0; j < KK; ++j) s += s_l1[tid * KK + j];
    s_l2[tid] = s;
  }
  __syncthreads();

  if (tid < L3N) {
    float s = 0.f;
#pragma unroll
    for (int j = 0; j < KK; ++j) s += s_l2[tid * KK + j];
    s_l3[tid] = s;
  }
  __syncthreads();

  // ---- Tail: loss, greedy descent, distance (tiny; thread 0) ----
  if (tid == 0) {
    float root = 0.f;
#pragma unroll
    for (int j = 0; j < L3N; ++j) root += s_l3[j];

    const int t = target[row];
    const float e_t = expf(s_row[t] - rmax);
    // win = (2^-1*S3 + 2^-2*S2 + 2^-3*S1 + (2^-4 + 2^-4)*e_t) / root
    const float num = 0.5f   * s_l3[t / 1000]
                    + 0.25f  * s_l2[t / 100]
                    + 0.125f * s_l1[t / 10]
                    + 0.125f * e_t;
    ws_loss[row] = logf(root) - logf(num);   // -log(win)

    // Greedy argmax descent, ties -> first (strict '>' keeps first).
    int n3 = 0;
    {
      float bv = s_l3[0];
      for (int j = 1; j < KK; ++j) if (s_l3[j] > bv) { bv = s_l3[j]; n3 = j; }
    }
    int n2 = n3 * KK;
    {
      float bv = s_l2[n2];
      for (int j = 1; j < KK; ++j)
        if (s_l2[n3 * KK + j] > bv) { bv = s_l2[n3 * KK + j]; n2 = n3 * KK + j; }
    }
    int n1 = n2 * KK;
    {
      float bv = s_l1[n1];
      for (int j = 1; j < KK; ++j)
        if (s_l1[n2 * KK + j] > bv) { bv = s_l1[n2 * KK + j]; n1 = n2 * KK + j; }
    }
    int lf = n1 * KK;   // leaf level: exp is monotone -> compare raw logits
    {
      float bv = s_row[lf];
      for (int j = 1; j < KK; ++j)
        if (s_row[n1 * KK + j] > bv) { bv = s_row[n1 * KK + j]; lf = n1 * KK + j; }
    }
    out[1 + row] = (float)lf;   // preds promoted to f32 by jnp.concatenate

    // dist = smallest m with pred//10^m == t//10^m (levels have disjoint ids), else D.
    int d;
    if      (lf == t)               d = 0;
    else if (lf / 10   == t / 10)   d = 1;
    else if (lf / 100  == t / 100)  d = 2;
    else if (lf / 1000 == t / 1000) d = 3;
    else                            d = DD;
    ws_dist[row] = (float)d;
  }
}

// Deterministic final reduction (fixed-order, no float atomics).
extern "C" __global__ void __launch_bounds__(NTHREADS)
hloss_reduce_kernel(const float* __restrict__ ws_loss,
                    const float* __restrict__ ws_dist,
                    float* __restrict__ out)
{
  __shared__ float s_a[NWAVES];
  __shared__ float s_b[NWAVES];
  const int tid = threadIdx.x;
  float a = 0.f, b = 0.f;
  for (int i = tid; i < BB; i += NTHREADS) {
    a += ws_loss[i];
    b += ws_dist[i];
  }
  a = wave_sum32(a);
  b = wave_sum32(b);
  if ((tid & 31) == 0) { s_a[tid >> 5] = a; s_b[tid >> 5] = b; }
  __syncthreads();
  if (tid == 0) {
    float ta = 0.f, tb = 0.f;
#pragma unroll
    for (int w = 0; w < NWAVES; ++w) { ta += s_a[w]; tb += s_b[w]; }
    out[0]      = ta / (float)BB;   // mean loss
    out[1 + BB] = tb;               // total_dist
  }
}

extern "C" void kernel_launch(void* const* d_in, const int* in_sizes, int n_in,
                              void* d_out, int out_size, void* d_ws, size_t ws_size,
                              hipStream_t stream) {
  const float* outputs = (const float*)d_in[0];  // [B, L] f32
  const int*   target  = (const int*)d_in[1];    // [B] i32
  // d_in[2] (chain) and d_in[3] (children) are implied by the balanced tree.
  float* out     = (float*)d_out;                // 2050 floats
  float* ws_loss = (float*)d_ws;                 // [B]
  float* ws_dist = ws_loss + BB;                 // [B]

  hloss_row_kernel<<<BB, NTHREADS, 0, stream>>>(outputs, target, out, ws_loss, ws_dist);
  hloss_reduce_kernel<<<1, NTHREADS, 0, stream>>>(ws_loss, ws_dist, out);
}